// GNOEncoder_6768868458576
// MI455X (gfx1250) — compile-verified
//
#include <hip/hip_runtime.h>

typedef __attribute__((ext_vector_type(2))) float v2f;
typedef __attribute__((ext_vector_type(8))) float v8f;

// Problem constants (match reference setup_inputs)
static constexpr int Bb  = 2;
static constexpr int Nn  = 1024;
static constexpr int Mm  = 1024;
static constexpr int DIN = 16;
static constexpr int Cc  = 32;
static constexpr int Hh  = 32;

// -------------------------------------------------------------------------
// Kernel 0: per-(b,d) rescale of coords to [-1,1]
// -------------------------------------------------------------------------
__global__ void rescale_kernel(const float* __restrict__ x, float* __restrict__ xs) {
    __shared__ float smn[256], smx[256];
    const int bd = blockIdx.x;          // 0..3 => (b,d)
    const int b = bd >> 1, d = bd & 1;
    const int t = threadIdx.x;
    float v[4];
    float mn = 1e30f, mx = -1e30f;
#pragma unroll
    for (int i = 0; i < 4; ++i) {
        const int n = t + i * 256;
        v[i] = x[(b * Nn + n) * 2 + d];
        mn = fminf(mn, v[i]);
        mx = fmaxf(mx, v[i]);
    }
    smn[t] = mn; smx[t] = mx;
    __syncthreads();
    for (int s = 128; s > 0; s >>= 1) {
        if (t < s) {
            smn[t] = fminf(smn[t], smn[t + s]);
            smx[t] = fmaxf(smx[t], smx[t + s]);
        }
        __syncthreads();
    }
    mn = smn[0]; mx = smx[0];
    const float sc = 2.0f / (mx - mn + 1e-12f);
#pragma unroll
    for (int i = 0; i < 4; ++i) {
        const int n = t + i * 256;
        xs[(b * Nn + n) * 2 + d] = (v[i] - mn) * sc - 1.0f;
    }
}

// -------------------------------------------------------------------------
// Kernel 1: lifting f = pndata @ W_lift + b_lift  via fp32 WMMA (K=16 -> 4 steps)
// One wave per 16x16 output tile: 128 row-tiles x 2 col-tiles = 256 waves.
// -------------------------------------------------------------------------
__global__ void __launch_bounds__(128) lift_kernel(const float* __restrict__ pnd,
                                                   const float* __restrict__ WL,
                                                   const float* __restrict__ bl,
                                                   float* __restrict__ f) {
    const int wid  = (blockIdx.x * blockDim.x + threadIdx.x) >> 5;  // 0..255
    const int lane = threadIdx.x & 31;
    const int rt = wid >> 1, ct = wid & 1;
    const int row0 = rt * 16;
    const int lrow = lane & 15;
    const int hsel = lane >> 4;          // 0: K pair {0,1}; 1: K pair {2,3}
    const int c = ct * 16 + lrow;

    v8f acc = {0.f, 0.f, 0.f, 0.f, 0.f, 0.f, 0.f, 0.f};
#pragma unroll
    for (int kk = 0; kk < 4; ++kk) {
        const int k0 = kk * 4 + 2 * hsel;
        v2f a, bm;
        // A (16x4 fp32 layout): lane row = lrow, K pair (k0, k0+1)
        a.x = pnd[(row0 + lrow) * DIN + k0];
        a.y = pnd[(row0 + lrow) * DIN + k0 + 1];
        // B (4x16 fp32 layout): lanes 0-15 rows k0,k0+1 ; lanes 16-31 rows k0+2,k0+3
        bm.x = WL[k0 * Cc + c];
        bm.y = WL[(k0 + 1) * Cc + c];
        acc = __builtin_amdgcn_wmma_f32_16x16x4_f32(false, a, false, bm,
                                                    (short)0, acc, false, false);
    }
    const float bias = bl[c];
#pragma unroll
    for (int r = 0; r < 8; ++r) {
        // C/D layout: lanes 0-15 -> row r ; lanes 16-31 -> row r+8 ; col = c
        f[(row0 + r + 8 * hsel) * Cc + c] = acc[r] + bias;
    }
}

// -------------------------------------------------------------------------
// Kernel 2: neighbor counts per (b,m) for both radii -> inverse counts
// -------------------------------------------------------------------------
__global__ void counts_kernel(const float* __restrict__ xs, const float* __restrict__ lq,
                              float* __restrict__ inv) {
    const int wid  = (blockIdx.x * blockDim.x + threadIdx.x) >> 5;  // 0..2047
    const int lane = threadIdx.x & 31;
    const int b = wid / Mm, m = wid % Mm;
    const float q0 = lq[m * 2], q1 = lq[m * 2 + 1];
    float c1 = 0.f, c2 = 0.f;
    for (int n = lane; n < Nn; n += 32) {
        const float x0 = xs[(b * Nn + n) * 2];
        const float x1 = xs[(b * Nn + n) * 2 + 1];
        const float dx = q0 - x0, dy = q1 - x1;
        const float d2 = dx * dx + dy * dy;
        c1 += (d2 <= 0.01f) ? 1.f : 0.f;   // (0.1*1)^2
        c2 += (d2 <= 0.04f) ? 1.f : 0.f;   // (0.1*2)^2
    }
#pragma unroll
    for (int s = 16; s >= 1; s >>= 1) {
        c1 += __shfl_xor(c1, s, 32);
        c2 += __shfl_xor(c2, s, 32);
    }
    if (lane == 0) {
        inv[wid * 2]     = 1.0f / fmaxf(c1, 1.0f);
        inv[wid * 2 + 1] = 1.0f / fmaxf(c2, 1.0f);
    }
}

// -------------------------------------------------------------------------
// Kernel 3: main GNO integral transform.
// Block = 256 threads = 8 waves, one m per wave, all same batch b.
// Entire f[b] (128KB) + xs[b] (8KB) staged once into CDNA5's 320KB LDS;
// per 16-point n-tile: gelu MLP layer-1 in VALU (A-operand layout),
// 32x32 layer-2 via 8x2 V_WMMA_F32_16X16X4_F32, masked-mean drain from LDS.
// -------------------------------------------------------------------------
__device__ __forceinline__ float gelu_tanh(float x) {
    // jax.nn.gelu (approximate=True)
    const float u = x * x;
    const float t = 0.7978845608028654f * x * fmaf(0.044715f, u, 1.0f);
    const float e = __expf(-2.0f * t);
    const float th = (1.0f - e) * __frcp_rn(1.0f + e);
    return 0.5f * x * (1.0f + th);
}

__global__ void __launch_bounds__(256) main_kernel(
    const float* __restrict__ xs, const float* __restrict__ f,
    const float* __restrict__ inv, const float* __restrict__ lq,
    const float* __restrict__ W1, const float* __restrict__ b1,
    const float* __restrict__ W2, const float* __restrict__ b2,
    float* __restrict__ out) {
    extern __shared__ float smem[];          // N*C + N*2 floats = 136 KB
    float* fs  = smem;                       // f[b]  : 32768 floats
    float* xss = smem + Nn * Cc;             // xs[b] : 2048 floats

    const int tid  = threadIdx.x;
    const int b    = blockIdx.x / (Mm / 8);  // 8 m's per block
    const int mg   = blockIdx.x % (Mm / 8);
    const int wv   = tid >> 5;
    const int lane = tid & 31;
    const int m    = mg * 8 + wv;
    const int wid  = b * Mm + m;
    const int lrow = lane & 15;
    const int hsel = lane >> 4;

    // --- cooperative LDS staging (vectorized b128 traffic) ---
    {
        const float4* src = (const float4*)(f + b * Nn * Cc);
        float4* dst = (float4*)fs;
#pragma unroll
        for (int i = 0; i < (Nn * Cc / 4) / 256; ++i)     // 32 iters
            dst[tid + i * 256] = src[tid + i * 256];
        const float4* s2 = (const float4*)(xs + b * Nn * 2);
        float4* d2v = (float4*)xss;
#pragma unroll
        for (int i = 0; i < (Nn * 2 / 4) / 256; ++i)      // 2 iters
            d2v[tid + i * 256] = s2[tid + i * 256];
    }
    __syncthreads();

    const float q0 = lq[m * 2], q1 = lq[m * 2 + 1];
    const float inv1 = inv[wid * 2], inv2 = inv[wid * 2 + 1];

    // Per-lane layer-1 weights for the 16 h-columns this lane produces:
    // A-operand needs, at K-step kk, h = 4*kk + 2*hsel + {0,1}.
    float W1x[16], W1y[16], Cm[16];
#pragma unroll
    for (int j = 0; j < 16; ++j) {
        const int h = 4 * (j >> 1) + 2 * hsel + (j & 1);
        W1x[j] = W1[h];              // W1[x-coord 0, h]
        W1y[j] = W1[Hh + h];         // W1[x-coord 1, h]
        Cm[j]  = fmaf(q0, W1[2 * Hh + h], fmaf(q1, W1[3 * Hh + h], b1[h]));
    }

    // W2 as B-operands (4x16 fp32 layout), preloaded: 8 K-steps x 2 C-halves.
    v2f Bw0[8], Bw1[8];
#pragma unroll
    for (int kk = 0; kk < 8; ++kk) {
        const int k0 = 4 * kk + 2 * hsel;
        Bw0[kk].x = W2[k0 * Cc + lrow];
        Bw0[kk].y = W2[(k0 + 1) * Cc + lrow];
        Bw1[kk].x = W2[k0 * Cc + 16 + lrow];
        Bw1[kk].y = W2[(k0 + 1) * Cc + 16 + lrow];
    }
    const float bz0 = b2[lrow], bz1 = b2[16 + lrow];

    float o0 = 0.f, o1 = 0.f;
    for (int nt = 0; nt < Nn / 16; ++nt) {
        const int n = nt * 16 + lrow;
        const float x0 = xss[n * 2], x1 = xss[n * 2 + 1];
        const float dx = q0 - x0, dy = q1 - x1;
        const float d2 = dx * dx + dy * dy;
        const float w = ((d2 <= 0.01f) ? inv1 : 0.f) + ((d2 <= 0.04f) ? inv2 : 0.f);

        v8f acc0 = {bz0, bz0, bz0, bz0, bz0, bz0, bz0, bz0};
        v8f acc1 = {bz1, bz1, bz1, bz1, bz1, bz1, bz1, bz1};
#pragma unroll
        for (int kk = 0; kk < 8; ++kk) {
            v2f a;
            const float p0 = fmaf(x0, W1x[2 * kk],     fmaf(x1, W1y[2 * kk],     Cm[2 * kk]));
            const float p1 = fmaf(x0, W1x[2 * kk + 1], fmaf(x1, W1y[2 * kk + 1], Cm[2 * kk + 1]));
            a.x = gelu_tanh(p0);
            a.y = gelu_tanh(p1);
            acc0 = __builtin_amdgcn_wmma_f32_16x16x4_f32(false, a, false, Bw0[kk],
                                                         (short)0, acc0, false, false);
            acc1 = __builtin_amdgcn_wmma_f32_16x16x4_f32(false, a, false, Bw1[kk],
                                                         (short)0, acc1, false, false);
        }
        // Drain in C/D layout: vgpr r -> row r (+8 for lanes 16-31); col = lrow (+16).
        // f comes from LDS: 16 consecutive dwords per half-wave, lanes 16-31 read the row 8 below.
#pragma unroll
        for (int r = 0; r < 8; ++r) {
            const int row = r + 8 * hsel;
            const float wr = __shfl(w, row, 32);      // w computed at lane == row
            const int nrow = nt * 16 + row;
            const float f0 = fs[nrow * Cc + lrow];
            const float f1 = fs[nrow * Cc + 16 + lrow];
            o0 = fmaf(acc0[r] * wr, f0, o0);
            o1 = fmaf(acc1[r] * wr, f1, o1);
        }
    }
    // Combine the two row halves (lanes l and l+16 hold disjoint row ranges).
    o0 += __shfl_xor(o0, 16, 32);
    o1 += __shfl_xor(o1, 16, 32);
    if (lane < 16) {
        out[wid * Cc + lane]      = o0;
        out[wid * Cc + 16 + lane] = o1;
    }
}

// -------------------------------------------------------------------------
// Host-side launcher
// -------------------------------------------------------------------------
extern "C" void kernel_launch(void* const* d_in, const int* in_sizes, int n_in,
                              void* d_out, int out_size, void* d_ws, size_t ws_size,
                              hipStream_t stream) {
    (void)in_sizes; (void)n_in; (void)out_size; (void)ws_size;
    const float* pndata = (const float*)d_in[0];   // [B,N,16]
    const float* xcoord = (const float*)d_in[1];   // [B,N,2]
    const float* lq     = (const float*)d_in[2];   // [M,2]
    const float* W_lift = (const float*)d_in[3];   // [16,32]
    const float* b_lift = (const float*)d_in[4];   // [32]
    const float* W1     = (const float*)d_in[5];   // [4,32]
    const float* b1     = (const float*)d_in[6];   // [32]
    const float* W2     = (const float*)d_in[7];   // [32,32]
    const float* b2     = (const float*)d_in[8];   // [32]
    float* out = (float*)d_out;                    // [B,M,32]

    float* ws  = (float*)d_ws;
    float* xs  = ws;                      // B*N*2   = 4096 floats (16B-aligned base)
    float* f   = xs + Bb * Nn * 2;        // B*N*32  = 65536 floats
    float* inv = f + Bb * Nn * Cc;        // B*M*2   = 4096 floats

    rescale_kernel<<<Bb * 2, 256, 0, stream>>>(xcoord, xs);
    lift_kernel<<<64, 128, 0, stream>>>(pndata, W_lift, b_lift, f);      // 256 waves
    counts_kernel<<<256, 256, 0, stream>>>(xs, lq, inv);                 // 2048 waves

    const size_t smem_bytes = (size_t)(Nn * Cc + Nn * 2) * sizeof(float); // 136 KB
    main_kernel<<<Bb * (Mm / 8), 256, smem_bytes, stream>>>(xs, f, inv, lq,
                                                            W1, b1, W2, b2, out);
}